// ContrastiveLossOriginal_62689342652555
// MI455X (gfx1250) — compile-verified
//
#include <hip/hip_runtime.h>

typedef __attribute__((ext_vector_type(2))) float v2f;
typedef __attribute__((ext_vector_type(8))) float v8f;

#define B_ROWS   4096
#define DIM      256
#define NROWS    8192
#define NTILES   (NROWS / 16)
#define TEMP_INV 1000.0f
#define LDS_STRIDE 260   // 256 + 4 pad -> conflict-free 8B ds reads

__device__ __forceinline__ const float* row_ptr(const float* p1, const float* p2, int row) {
    return (row < B_ROWS) ? (p1 + (size_t)row * DIM)
                          : (p2 + (size_t)(row - B_ROWS) * DIM);
}

// ---------------------------------------------------------------------------
// Kernel 1: rnorm[r] = 1 / max(||reps[r]||, 1e-12), one wave32 per row
// ---------------------------------------------------------------------------
__global__ void rnorm_kernel(const float* __restrict__ p1, const float* __restrict__ p2,
                             float* __restrict__ rnorm) {
    const int wave = threadIdx.x >> 5;
    const int lane = threadIdx.x & 31;
    const int row  = blockIdx.x * 8 + wave;
    const float4* rp4 = (const float4*)row_ptr(p1, p2, row);
    float4 a = rp4[lane * 2 + 0];
    float4 b = rp4[lane * 2 + 1];
    float ss = a.x*a.x + a.y*a.y + a.z*a.z + a.w*a.w
             + b.x*b.x + b.y*b.y + b.z*b.z + b.w*b.w;
    #pragma unroll
    for (int off = 16; off > 0; off >>= 1)
        ss += __shfl_xor(ss, off, 32);
    if (lane == 0) {
        float n = sqrtf(ss);
        rnorm[row] = 1.0f / fmaxf(n, 1e-12f);
    }
}

// ---------------------------------------------------------------------------
// Kernel 2: pos[k] = <z_i[k], z_j[k]> = (p1[k].p2[k]) * rnorm[k] * rnorm[B+k]
// ---------------------------------------------------------------------------
__global__ void pos_kernel(const float* __restrict__ p1, const float* __restrict__ p2,
                           const float* __restrict__ rnorm, float* __restrict__ pos) {
    const int wave = threadIdx.x >> 5;
    const int lane = threadIdx.x & 31;
    const int k    = blockIdx.x * 8 + wave;
    const float4* a4 = (const float4*)(p1 + (size_t)k * DIM);
    const float4* b4 = (const float4*)(p2 + (size_t)k * DIM);
    float d = 0.0f;
    #pragma unroll
    for (int q = 0; q < 2; ++q) {
        float4 a = a4[lane * 2 + q];
        float4 b = b4[lane * 2 + q];
        d += a.x*b.x + a.y*b.y + a.z*b.z + a.w*b.w;
    }
    #pragma unroll
    for (int off = 16; off > 0; off >>= 1)
        d += __shfl_xor(d, off, 32);
    if (lane == 0)
        pos[k] = d * rnorm[k] * rnorm[B_ROWS + k];
}

// ---------------------------------------------------------------------------
// Kernel 3: fused sim = Z.Z^T (fp32 WMMA) + row logsumexp.
// 4 waves/block; wave owns rows [rowBase, rowBase+16); loops 512 column tiles.
// Column tiles are staged into double-buffered LDS with
// global_load_async_to_lds_b128 (ASYNCcnt), overlapping the copy of tile ct+1
// with the 64 fp32 WMMAs of tile ct. Two independent WMMA accumulator chains
// (even/odd K-steps) break the per-instruction D->C dependency.
// lse[r] = 1000 + log( sum_c exp((sim[r,c]-1)*1000) )  (row max is diag == 1)
// ---------------------------------------------------------------------------
#define ASYNC_B128(ldsoff, goff, base, IMM)                                     \
    asm volatile("global_load_async_to_lds_b128 %0, %1, %2 offset:" #IMM        \
                 :: "v"(ldsoff), "v"(goff), "s"(base) : "memory")

__launch_bounds__(128)
__global__ void simlse_kernel(const float* __restrict__ p1, const float* __restrict__ p2,
                              const float* __restrict__ rnorm, float* __restrict__ lse) {
    __shared__ __align__(16) float tile[2][16 * LDS_STRIDE];

    const int tid  = threadIdx.x;
    const int wave = tid >> 5;
    const int lane = tid & 31;
    const int m    = lane & 15;
    const int h    = lane >> 4;
    const int rowBase = blockIdx.x * 64 + wave * 16;

    // Register-resident A fragments: 64 K-steps of K=4 (row = rowBase+m)
    const float* arow = row_ptr(p1, p2, rowBase + m);
    v2f afrag[64];
    #pragma unroll
    for (int k4 = 0; k4 < 64; ++k4)
        afrag[k4] = *(const v2f*)(arow + 4 * k4 + 2 * h);

    // Row inverse-norms for this lane's 8 C rows (C row = v + 8*h)
    float rnA[8];
    #pragma unroll
    for (int v = 0; v < 8; ++v)
        rnA[v] = rnorm[rowBase + v + 8 * h];

    float accExp[8] = {0, 0, 0, 0, 0, 0, 0, 0};

    // --- async staging addresses -------------------------------------------
    // Each thread copies one 128B row segment: row trow, bytes [tq16, tq16+128)
    const int trow = tid >> 3;               // 0..15 : tile row
    const unsigned tq16 = (unsigned)(tid & 7) * 128u;  // byte base within row
    const unsigned goff = (unsigned)trow * (DIM * 4u) + tq16;  // global byte off
    unsigned ldsoff[2];
    ldsoff[0] = (unsigned)(size_t)(&tile[0][trow * LDS_STRIDE]) + tq16;
    ldsoff[1] = (unsigned)(size_t)(&tile[1][trow * LDS_STRIDE]) + tq16;

    // stage column tile ct into LDS buffer b (8 x 16B async copies per thread)
    auto stage = [&](int ct, int b) {
        const int row0 = ct * 16;       // tiles never straddle the p1/p2 split
        const float* base = (row0 < B_ROWS) ? (p1 + (size_t)row0 * DIM)
                                            : (p2 + (size_t)(row0 - B_ROWS) * DIM);
        const unsigned lo = ldsoff[b];
        ASYNC_B128(lo, goff, base, 0);
        ASYNC_B128(lo, goff, base, 16);
        ASYNC_B128(lo, goff, base, 32);
        ASYNC_B128(lo, goff, base, 48);
        ASYNC_B128(lo, goff, base, 64);
        ASYNC_B128(lo, goff, base, 80);
        ASYNC_B128(lo, goff, base, 96);
        ASYNC_B128(lo, goff, base, 112);
    };

    stage(0, 0);
    asm volatile("s_wait_asynccnt 0x0" ::: "memory");
    __syncthreads();

    for (int ct = 0; ct < NTILES; ++ct) {
        const int cur = ct & 1;
        if (ct + 1 < NTILES)
            stage(ct + 1, cur ^ 1);      // overlaps with compute below

        const float rnC = rnorm[ct * 16 + m];

        v8f c0 = {0, 0, 0, 0, 0, 0, 0, 0};
        v8f c1 = {0, 0, 0, 0, 0, 0, 0, 0};
        const float* brow = &tile[cur][m * LDS_STRIDE + 2 * h];
        #pragma unroll
        for (int k4 = 0; k4 < 64; k4 += 2) {
            v2f b0 = *(const v2f*)(brow + 4 * k4);
            v2f b1 = *(const v2f*)(brow + 4 * k4 + 4);
            c0 = __builtin_amdgcn_wmma_f32_16x16x4_f32(
                     false, afrag[k4],     false, b0, (short)0, c0, false, false);
            c1 = __builtin_amdgcn_wmma_f32_16x16x4_f32(
                     false, afrag[k4 + 1], false, b1, (short)0, c1, false, false);
        }

        #pragma unroll
        for (int v = 0; v < 8; ++v) {
            float s = (c0[v] + c1[v]) * rnA[v] * rnC;   // cosine similarity
            accExp[v] += __expf((s - 1.0f) * TEMP_INV);
        }

        asm volatile("s_wait_asynccnt 0x0" ::: "memory");  // next buffer ready
        __syncthreads();                                   // all waves done w/ cur
    }

    // Reduce each row-sum across the 16 lanes of this half-wave (cols of tile)
    #pragma unroll
    for (int v = 0; v < 8; ++v) {
        float a = accExp[v];
        #pragma unroll
        for (int off = 1; off < 16; off <<= 1)
            a += __shfl_xor(a, off, 32);
        accExp[v] = a;
    }
    if (m == 0) {
        #pragma unroll
        for (int v = 0; v < 8; ++v)
            lse[rowBase + v + 8 * h] = TEMP_INV + logf(accExp[v]);
    }
}

// ---------------------------------------------------------------------------
// Kernel 4: deterministic final reduction -> (loss, sum(positives))
// loss = (sum_r lse[r] - 2000 * sum_k pos[k]) / 8192 ; out[1] = 2*sum pos
// ---------------------------------------------------------------------------
__global__ void finalize_kernel(const float* __restrict__ lse, const float* __restrict__ pos,
                                float* __restrict__ out) {
    __shared__ float red[256];
    const int tid = threadIdx.x;
    float sl = 0.0f, sp = 0.0f;
    for (int i = tid; i < NROWS; i += 256)  sl += lse[i];
    for (int i = tid; i < B_ROWS; i += 256) sp += pos[i];

    red[tid] = sl;
    __syncthreads();
    for (int s = 128; s > 0; s >>= 1) {
        if (tid < s) red[tid] += red[tid + s];
        __syncthreads();
    }
    const float totL = red[0];
    __syncthreads();

    red[tid] = sp;
    __syncthreads();
    for (int s = 128; s > 0; s >>= 1) {
        if (tid < s) red[tid] += red[tid + s];
        __syncthreads();
    }
    if (tid == 0) {
        const float totP = red[0];
        out[0] = (totL - 2.0f * TEMP_INV * totP) / (float)NROWS;
        out[1] = 2.0f * totP;
    }
}

// ---------------------------------------------------------------------------
extern "C" void kernel_launch(void* const* d_in, const int* in_sizes, int n_in,
                              void* d_out, int out_size, void* d_ws, size_t ws_size,
                              hipStream_t stream) {
    (void)in_sizes; (void)n_in; (void)out_size; (void)ws_size;
    const float* p1 = (const float*)d_in[0];
    const float* p2 = (const float*)d_in[1];
    float* out = (float*)d_out;

    float* ws    = (float*)d_ws;
    float* rnorm = ws;                       // 8192 floats
    float* pos   = ws + NROWS;               // 4096 floats
    float* lse   = ws + NROWS + B_ROWS;      // 8192 floats

    rnorm_kernel   <<<NROWS / 8, 256, 0, stream>>>(p1, p2, rnorm);
    pos_kernel     <<<B_ROWS / 8, 256, 0, stream>>>(p1, p2, rnorm, pos);
    simlse_kernel  <<<NROWS / 64, 128, 0, stream>>>(p1, p2, rnorm, lse);
    finalize_kernel<<<1, 256, 0, stream>>>(lse, pos, out);
}